// RetrieveModule_40836549050719
// MI455X (gfx1250) — compile-verified
//
#include <hip/hip_runtime.h>
#include <math.h>

// Problem dims (fixed by harness)
#define B_TOT   1024
#define N_CAND  131072
#define D_DIM   256
#define DB_DIM  512
#define C_CTX   96
#define CHUNK   256     // batch rows per score-buffer chunk

typedef float v2f __attribute__((ext_vector_type(2)));
typedef float v4f __attribute__((ext_vector_type(4)));
typedef float v8f __attribute__((ext_vector_type(8)));

__device__ __forceinline__ v8f wmma_f32(v2f a, v2f b, v8f c) {
  // D = A(16x4) * B(4x16) + C, fp32
  return __builtin_amdgcn_wmma_f32_16x16x4_f32(false, a, false, b, (short)0, c, false, false);
}

__device__ __forceinline__ unsigned fkey(float f) {
  unsigned u = __float_as_uint(f);
  return (u & 0x80000000u) ? ~u : (u | 0x80000000u);
}

// gfx1250 async global->LDS copy (16B per lane), tracked with ASYNCcnt.
// VDST VGPR carries the LDS byte offset, VADDR the 64-bit global address (GV).
__device__ __forceinline__ void async_copy_b128(void* lds_dst, const void* gsrc) {
  unsigned lds_off = (unsigned)(uintptr_t)lds_dst;  // low 32 bits = LDS offset
  asm volatile("global_load_async_to_lds_b128 %0, %1, off"
               :: "v"(lds_off), "v"(gsrc) : "memory");
}
__device__ __forceinline__ void async_wait_all() {
  asm volatile("s_wait_asynccnt 0" ::: "memory");
}

// ---------------------------------------------------------------------------
// Kernel 0: candidate squared norms. One 256-thread block per candidate row.
// ---------------------------------------------------------------------------
__global__ __launch_bounds__(256) void cnorm_kernel(const float* __restrict__ ck,
                                                    float* __restrict__ cn) {
  __shared__ float red[256];
  const int row = blockIdx.x;
  float v = ck[(size_t)row * D_DIM + threadIdx.x];
  red[threadIdx.x] = v * v;
  __syncthreads();
  for (int s = 128; s > 0; s >>= 1) {
    if (threadIdx.x < s) red[threadIdx.x] += red[threadIdx.x + s];
    __syncthreads();
  }
  if (threadIdx.x == 0) cn[row] = red[0];
}

// ---------------------------------------------------------------------------
// Kernel 1: scores[m, n] = 2 * (k[m] . ck[n]) - ||ck[n]||^2
// Block: 256 threads (8 waves). Tile 64(M) x 128(N), K = 256 in steps of 4.
// LDS staging via async direct-to-LDS B128 copies (no VGPR bounce), stride 260
// padding (==4 mod 64 banks -> conflict-free WMMA operand reads).
// ---------------------------------------------------------------------------
#define TM  64
#define TN  128
#define LDA 260

__global__ __launch_bounds__(256)
void score_gemm_kernel(const float* __restrict__ K,   // [CHUNK, D]
                       const float* __restrict__ CK,  // [N, D]
                       const float* __restrict__ cn,  // [N]
                       float* __restrict__ scores) {  // [CHUNK, N]
  extern __shared__ char smem_raw[];
  float* As = (float*)smem_raw;        // [TM][LDA]
  float* Bs = As + TM * LDA;           // [TN][LDA]

  const int tid   = threadIdx.x;
  const int lane  = tid & 31;
  const int wave  = tid >> 5;
  const int m_blk = blockIdx.y * TM;
  const int n_blk = blockIdx.x * TN;

  // Async staging: each iteration moves 16B/lane straight into LDS.
  for (int i = tid; i < TM * D_DIM / 4; i += 256) {
    const int f = i * 4, r = f / D_DIM, c = f % D_DIM;
    async_copy_b128(As + r * LDA + c, K + (size_t)(m_blk + r) * D_DIM + c);
  }
  for (int i = tid; i < TN * D_DIM / 4; i += 256) {
    const int f = i * 4, r = f / D_DIM, c = f % D_DIM;
    async_copy_b128(Bs + r * LDA + c, CK + (size_t)(n_blk + r) * D_DIM + c);
  }
  async_wait_all();
  __syncthreads();

  const int wm  = (wave & 3) * 16;   // 0,16,32,48
  const int wn  = (wave >> 2) * 64;  // 0 or 64
  const int hl  = lane >> 4;         // lane half: selects K pair / M half
  const int l16 = lane & 15;

  v8f acc[4];
  #pragma unroll
  for (int t = 0; t < 4; ++t)
    #pragma unroll
    for (int r = 0; r < 8; ++r) acc[t][r] = 0.0f;

  const float* Arow = As + (wm + l16) * LDA + 2 * hl;
  #pragma unroll 8
  for (int k0 = 0; k0 < D_DIM; k0 += 4) {
    v2f a = *(const v2f*)(Arow + k0);
    #pragma unroll
    for (int t = 0; t < 4; ++t) {
      v2f b = *(const v2f*)(Bs + (wn + t * 16 + l16) * LDA + k0 + 2 * hl);
      acc[t] = wmma_f32(a, b, acc[t]);
    }
  }

  // Epilogue: lane covers n = l16 column, rows m = r + 8*hl.
  #pragma unroll
  for (int t = 0; t < 4; ++t) {
    const int n   = n_blk + wn + t * 16 + l16;
    const float c = cn[n];
    #pragma unroll
    for (int r = 0; r < 8; ++r) {
      const int m = m_blk + wm + r + 8 * hl;
      scores[(size_t)m * N_CAND + n] = 2.0f * acc[t][r] - c;
    }
  }
}

// ---------------------------------------------------------------------------
// Kernel 2: exact top-96 per row via 12-bit radix histogram + exact tie pass.
// One 256-thread block per row of the chunk.
// ---------------------------------------------------------------------------
#define NBINS  4096
#define TIECAP 4096

__global__ __launch_bounds__(256)
void select_kernel(const float* __restrict__ scores, int row_base,
                   int* __restrict__ ctx_idx) {
  __shared__ unsigned hist[NBINS];
  __shared__ unsigned tieKey[TIECAP];
  __shared__ int      tieIdx[TIECAP];
  __shared__ int      ctxL[C_CTX];
  __shared__ unsigned chunkSum[256];
  __shared__ unsigned bestK[256];
  __shared__ int      bestI[256];
  __shared__ int      bestP[256];
  __shared__ int      sT, topCnt, tieCnt;

  const int tid = threadIdx.x;
  const float* srow = scores + (size_t)blockIdx.x * N_CAND;

  for (int i = tid; i < NBINS; i += 256) hist[i] = 0u;
  if (tid == 0) { topCnt = 0; tieCnt = 0; }
  __syncthreads();

  for (int n = tid; n < N_CAND; n += 256) {
    unsigned u = fkey(srow[n]);
    atomicAdd(&hist[u >> 20], 1u);
  }
  __syncthreads();

  unsigned cs = 0;
  for (int i = 0; i < 16; ++i) cs += hist[tid * 16 + i];
  chunkSum[tid] = cs;
  __syncthreads();

  if (tid == 0) {
    unsigned acc = 0;
    int tc = 0;
    for (int t = 255; t >= 0; --t) {
      if (acc + chunkSum[t] >= (unsigned)C_CTX) { tc = t; break; }
      acc += chunkSum[t];
    }
    int T = tc * 16;
    for (int bn = tc * 16 + 15; bn >= tc * 16; --bn) {
      if (acc + hist[bn] >= (unsigned)C_CTX) { T = bn; break; }
      acc += hist[bn];
    }
    sT = T;
  }
  __syncthreads();

  const int T = sT;
  for (int n = tid; n < N_CAND; n += 256) {
    unsigned u = fkey(srow[n]);
    int bn = (int)(u >> 20);
    if (bn > T) {
      int p = atomicAdd(&topCnt, 1);
      if (p < C_CTX) ctxL[p] = n;
    } else if (bn == T) {
      int p = atomicAdd(&tieCnt, 1);
      if (p < TIECAP) { tieKey[p] = u; tieIdx[p] = n; }
    }
  }
  __syncthreads();

  int G = topCnt; if (G > C_CTX) G = C_CTX;
  const int Tn = (tieCnt < TIECAP) ? tieCnt : TIECAP;
  const int R  = C_CTX - G;

  for (int r = 0; r < R; ++r) {
    unsigned bk = 0u; int bi = 0x7fffffff, bp = -1;
    for (int i = tid; i < Tn; i += 256) {
      unsigned kk = tieKey[i];
      if (kk > bk || (kk == bk && tieIdx[i] < bi)) { bk = kk; bi = tieIdx[i]; bp = i; }
    }
    bestK[tid] = bk; bestI[tid] = bi; bestP[tid] = bp;
    __syncthreads();
    for (int s = 128; s > 0; s >>= 1) {
      if (tid < s) {
        if (bestK[tid + s] > bestK[tid] ||
            (bestK[tid + s] == bestK[tid] && bestI[tid + s] < bestI[tid])) {
          bestK[tid] = bestK[tid + s];
          bestI[tid] = bestI[tid + s];
          bestP[tid] = bestP[tid + s];
        }
      }
      __syncthreads();
    }
    if (tid == 0) {
      ctxL[G + r] = (bestP[0] >= 0) ? bestI[0] : 0;
      if (bestP[0] >= 0) tieKey[bestP[0]] = 0u;
    }
    __syncthreads();
  }

  if (tid < C_CTX)
    ctx_idx[(size_t)(row_base + blockIdx.x) * C_CTX + tid] = ctxL[tid];
}

// ---------------------------------------------------------------------------
// Kernel 3: gather + MLP + softmax-weighted reduce. One block per batch row.
//   diff[c,d]=k-ck -> h=relu(diff@W1^T+b1) -> v=h@W2^T ; sim=-||diff||^2
//   out = x + b_lbl + (sum probs*cy)*W_lbl + probs @ v
// ---------------------------------------------------------------------------
#define LDD 260
#define LDH 132
#define ES  128   // e-strip width for DB=512

__global__ __launch_bounds__(256)
void mlp_kernel(const float* __restrict__ x,  const float* __restrict__ K,
                const float* __restrict__ CK, const float* __restrict__ cy_all,
                const float* __restrict__ W_lbl, const float* __restrict__ b_lbl,
                const float* __restrict__ W1, const float* __restrict__ b1,
                const float* __restrict__ W2,
                const int* __restrict__ ctx_idx, float* __restrict__ out) {
  extern __shared__ char smem_raw[];
  float* diff  = (float*)smem_raw;            // [96][LDD]
  float* hbuf  = diff + C_CTX * LDD;          // [96][LDH]
  float* kb    = hbuf + C_CTX * LDH;          // [256]
  float* sim   = kb + D_DIM;                  // [96]
  float* probs = sim + C_CTX;                 // [96]
  float* cyv   = probs + C_CTX;               // [96]
  float* red   = cyv + C_CTX;                 // [256]
  float* sc    = red + 256;                   // [1]

  const int b    = blockIdx.x;
  const int tid  = threadIdx.x;
  const int lane = tid & 31;
  const int wave = tid >> 5;

  kb[tid] = K[(size_t)b * D_DIM + tid];
  __syncthreads();

  // --- gather + diff + sim (one wave per context element) ---
  for (int c = wave; c < C_CTX; c += 8) {
    const int idx = ctx_idx[(size_t)b * C_CTX + c];
    const float* crow = CK + (size_t)idx * D_DIM;
    float ss = 0.0f;
    #pragma unroll
    for (int j = 0; j < 8; ++j) {
      const int d = lane + j * 32;
      const float df = kb[d] - crow[d];
      diff[c * LDD + d] = df;
      ss += df * df;
    }
    for (int o = 16; o > 0; o >>= 1) ss += __shfl_xor(ss, o, 32);
    if (lane == 0) { sim[c] = -ss; cyv[c] = cy_all[idx]; }
  }
  __syncthreads();

  // --- softmax over 96 + s_cy ---
  const float sv = (tid < C_CTX) ? sim[tid] : -3.4e38f;
  red[tid] = sv; __syncthreads();
  for (int s = 128; s > 0; s >>= 1) {
    if (tid < s) red[tid] = fmaxf(red[tid], red[tid + s]);
    __syncthreads();
  }
  const float mx = red[0]; __syncthreads();
  const float ev = (tid < C_CTX) ? __expf(sv - mx) : 0.0f;
  red[tid] = ev; __syncthreads();
  for (int s = 128; s > 0; s >>= 1) {
    if (tid < s) red[tid] += red[tid + s];
    __syncthreads();
  }
  const float denom = red[0]; __syncthreads();
  const float pv = ev / denom;
  if (tid < C_CTX) probs[tid] = pv;
  red[tid] = (tid < C_CTX) ? pv * cyv[tid] : 0.0f;
  __syncthreads();
  for (int s = 128; s > 0; s >>= 1) {
    if (tid < s) red[tid] += red[tid + s];
    __syncthreads();
  }
  if (tid == 0) sc[0] = red[0];
  __syncthreads();

  // --- strip-mined GEMM1 (+bias+relu) -> GEMM2 with persistent accumulators ---
  const int hl  = lane >> 4;
  const int l16 = lane & 15;

  v8f acc2[2][6];
  #pragma unroll
  for (int dc = 0; dc < 2; ++dc)
    #pragma unroll
    for (int mt = 0; mt < 6; ++mt)
      #pragma unroll
      for (int r = 0; r < 8; ++r) acc2[dc][mt][r] = 0.0f;

  for (int s = 0; s < DB_DIM / ES; ++s) {
    const int e0 = s * ES;

    // GEMM1: h[m, e0+e'] = relu(diff @ W1^T + b1), 48 tiles / 8 waves
    for (int j = 0; j < 6; ++j) {
      const int t  = wave + j * 8;
      const int mi = t % 6;
      const int ei = t / 6;
      v8f a1;
      #pragma unroll
      for (int r = 0; r < 8; ++r) a1[r] = 0.0f;
      const float* Aro = diff + (mi * 16 + l16) * LDD + 2 * hl;
      const float* Wro = W1 + (size_t)(e0 + ei * 16 + l16) * D_DIM + 2 * hl;
      #pragma unroll 8
      for (int k0 = 0; k0 < D_DIM; k0 += 4) {
        v2f av = *(const v2f*)(Aro + k0);
        v2f wv = *(const v2f*)(Wro + k0);
        a1 = wmma_f32(av, wv, a1);
      }
      const float bb = b1[e0 + ei * 16 + l16];
      #pragma unroll
      for (int r = 0; r < 8; ++r) {
        const float hv = fmaxf(a1[r] + bb, 0.0f);
        hbuf[(mi * 16 + r + 8 * hl) * LDH + ei * 16 + l16] = hv;
      }
    }
    __syncthreads();

    // GEMM2: values += h_strip @ W2_strip^T ; wave owns d-cols [wave*32, +32)
    #pragma unroll
    for (int dc = 0; dc < 2; ++dc) {
      const int dcol = (wave * 2 + dc) * 16;
      const float* W2ro = W2 + (size_t)(dcol + l16) * DB_DIM + e0 + 2 * hl;
      for (int mt = 0; mt < 6; ++mt) {
        const float* Hro = hbuf + (mt * 16 + l16) * LDH + 2 * hl;
        v8f a2 = acc2[dc][mt];
        #pragma unroll 8
        for (int k0 = 0; k0 < ES; k0 += 4) {
          v2f hv = *(const v2f*)(Hro + k0);
          v2f wv = *(const v2f*)(W2ro + k0);
          a2 = wmma_f32(hv, wv, a2);
        }
        acc2[dc][mt] = a2;
      }
    }
    __syncthreads();
  }

  // --- epilogue: out = x + b_lbl + s_cy*W_lbl + probs @ values ---
  const float s_cy = sc[0];
  #pragma unroll
  for (int dc = 0; dc < 2; ++dc) {
    const int d = (wave * 2 + dc) * 16 + l16;
    float part = 0.0f;
    #pragma unroll
    for (int mt = 0; mt < 6; ++mt)
      #pragma unroll
      for (int r = 0; r < 8; ++r)
        part += probs[mt * 16 + r + 8 * hl] * acc2[dc][mt][r];
    part += __shfl_xor(part, 16, 32);
    if (hl == 0)
      out[(size_t)b * D_DIM + d] =
          x[(size_t)b * D_DIM + d] + b_lbl[d] + s_cy * W_lbl[d] + part;
  }
}

// ---------------------------------------------------------------------------
extern "C" void kernel_launch(void* const* d_in, const int* in_sizes, int n_in,
                              void* d_out, int out_size, void* d_ws, size_t ws_size,
                              hipStream_t stream) {
  const float* x     = (const float*)d_in[0];
  const float* k     = (const float*)d_in[1];
  const float* ck    = (const float*)d_in[2];
  const float* cy    = (const float*)d_in[3];
  const float* W_lbl = (const float*)d_in[4];
  const float* b_lbl = (const float*)d_in[5];
  const float* W1    = (const float*)d_in[6];
  const float* b1    = (const float*)d_in[7];
  const float* W2    = (const float*)d_in[8];
  float* out = (float*)d_out;

  char* ws = (char*)d_ws;
  float* cn = (float*)ws;                     ws += (size_t)N_CAND * sizeof(float);
  int* ctx  = (int*)ws;                       ws += (size_t)B_TOT * C_CTX * sizeof(int);
  ws = (char*)(((uintptr_t)ws + 255) & ~(uintptr_t)255);
  float* scores = (float*)ws;                 // [CHUNK, N] reused per chunk

  cnorm_kernel<<<N_CAND, 256, 0, stream>>>(ck, cn);

  const size_t gemm_lds = (size_t)(TM * LDA + TN * LDA) * sizeof(float);
  for (int cb = 0; cb < B_TOT / CHUNK; ++cb) {
    score_gemm_kernel<<<dim3(N_CAND / TN, CHUNK / TM), 256, gemm_lds, stream>>>(
        k + (size_t)cb * CHUNK * D_DIM, ck, cn, scores);
    select_kernel<<<CHUNK, 256, 0, stream>>>(scores, cb * CHUNK, ctx);
  }

  const size_t mlp_lds =
      (size_t)(C_CTX * LDD + C_CTX * LDH + D_DIM + 3 * C_CTX + 256 + 8) * sizeof(float);
  mlp_kernel<<<B_TOT, 256, mlp_lds, stream>>>(x, k, ck, cy, W_lbl, b_lbl, W1, b1,
                                              W2, ctx, out);
}